// diaglinear_layer_16819091931190
// MI455X (gfx1250) — compile-verified
//
#include <hip/hip_runtime.h>

// out[r, c] = x[r, c] * weight[c]   (fp32, 32768 x 4096)
// Memory-bound streaming kernel: ~1.07 GB of traffic, ~46 us floor @ 23.3 TB/s.

typedef float v4f __attribute__((ext_vector_type(4)));
typedef int   v4i __attribute__((__vector_size__(4 * sizeof(int))));  // matches builtin param

typedef __attribute__((address_space(1))) v4i gv4i_t;  // global (printed as __device__)
typedef __attribute__((address_space(3))) v4i lv4i_t;  // LDS    (printed as __shared__)

#define KDIM   4096
#define KROWS  32768
#define KBLOCK 256                       // 8 wave32 per block
#define KSEGS  (KDIM / (KBLOCK * 4))     // 4 column segments of 1024 floats
#define KGRID  1024                      // blocks; each owns KROWS/KGRID rows
#define ROWS_PER_BLOCK (KROWS / KGRID)   // 32

static_assert(KDIM == KBLOCK * 4 * KSEGS, "column tiling must cover the row");
static_assert(KROWS % KGRID == 0, "rows must divide evenly");
static_assert(ROWS_PER_BLOCK % 2 == 0, "2-row unroll");

__global__ __launch_bounds__(KBLOCK) void diaglinear_scale_kernel(
    const float* __restrict__ x,
    const float* __restrict__ w,
    float* __restrict__ out)
{
    __shared__ float wlds[KDIM];   // 16 KB staged weight row
    const int tid = threadIdx.x;
    const int col0 = tid * 4;

    // ---- Stage weight (16 KB) into LDS once per block ----
    // CDNA5 async global->LDS DMA (ASYNCcnt-tracked) broadcast path.
#if __has_builtin(__builtin_amdgcn_global_load_async_to_lds_b128)
    #pragma unroll
    for (int s = 0; s < KSEGS; ++s) {
        const float* gp = w + s * (KBLOCK * 4) + col0;
        float*       lp = &wlds[s * (KBLOCK * 4) + col0];
        __builtin_amdgcn_global_load_async_to_lds_b128(
            (gv4i_t*)gp, (lv4i_t*)lp, /*offset=*/0, /*cpol=*/0);
    }
  #if __has_builtin(__builtin_amdgcn_s_wait_asynccnt)
    __builtin_amdgcn_s_wait_asynccnt(0);
  #else
    asm volatile("s_wait_asynccnt 0" ::: "memory");
  #endif
#else
    #pragma unroll
    for (int s = 0; s < KSEGS; ++s) {
        *(v4f*)&wlds[s * (KBLOCK * 4) + col0] =
            *(const v4f*)(w + s * (KBLOCK * 4) + col0);
    }
#endif
    __syncthreads();

    // Pull this thread's 16 weight values into registers (ds_load_b128 x4);
    // they are reused for all ROWS_PER_BLOCK rows.
    v4f wv[KSEGS];
    #pragma unroll
    for (int s = 0; s < KSEGS; ++s)
        wv[s] = *(const v4f*)&wlds[s * (KBLOCK * 4) + col0];

    // ---- Stream rows: 2-row unroll -> 8 independent b128 NT loads in flight ----
    const int r0 = blockIdx.x * ROWS_PER_BLOCK;
    #pragma unroll 1
    for (int r = r0; r < r0 + ROWS_PER_BLOCK; r += 2) {
        const size_t b0 = (size_t)r * KDIM + col0;
        const size_t b1 = b0 + KDIM;

        // Speculative prefetch two rows ahead (global_prefetch_b8); harmless
        // at the tail (dropped on translation failure per ISA).
        __builtin_prefetch(x + b0 + 2 * (size_t)KDIM, 0, 0);
        __builtin_prefetch(x + b1 + 2 * (size_t)KDIM, 0, 0);

        v4f a[KSEGS], b[KSEGS];
        #pragma unroll
        for (int s = 0; s < KSEGS; ++s)
            a[s] = __builtin_nontemporal_load((const v4f*)(x + b0 + s * (KBLOCK * 4)));
        #pragma unroll
        for (int s = 0; s < KSEGS; ++s)
            b[s] = __builtin_nontemporal_load((const v4f*)(x + b1 + s * (KBLOCK * 4)));

        #pragma unroll
        for (int s = 0; s < KSEGS; ++s)
            __builtin_nontemporal_store(a[s] * wv[s], (v4f*)(out + b0 + s * (KBLOCK * 4)));
        #pragma unroll
        for (int s = 0; s < KSEGS; ++s)
            __builtin_nontemporal_store(b[s] * wv[s], (v4f*)(out + b1 + s * (KBLOCK * 4)));
    }
}

extern "C" void kernel_launch(void* const* d_in, const int* in_sizes, int n_in,
                              void* d_out, int out_size, void* d_ws, size_t ws_size,
                              hipStream_t stream) {
    const float* x = (const float*)d_in[0];   // (32768, 4096) fp32
    const float* w = (const float*)d_in[1];   // (4096,)       fp32
    float* out = (float*)d_out;               // (32768, 4096) fp32
    (void)in_sizes; (void)n_in; (void)out_size; (void)d_ws; (void)ws_size;

    diaglinear_scale_kernel<<<KGRID, KBLOCK, 0, stream>>>(x, w, out);
}